// VerticalAttention_3788161155171
// MI455X (gfx1250) — compile-verified
//
#include <hip/hip_runtime.h>
#include <hip/hip_bf16.h>

// VerticalAttention for MI455X (gfx1250, wave32).
// Bandwidth-bound: 2 x 1GiB feature streams dominate (~90us @ 23.3TB/s).
// GEMM (4096x256x256) done with V_WMMA_F32_16X16X4_F32 (full fp32 path).

#define BB   32
#define CC   256
#define HH   128
#define WW   256
#define ATT  256
#define HID  256
#define MINW 32
#define WIN  8     // WW / MINW

typedef float f32x4 __attribute__((ext_vector_type(4)));
typedef float v2f   __attribute__((ext_vector_type(2)));
typedef float v8f   __attribute__((ext_vector_type(8)));

// ---------------------------------------------------------------------------
// Kernel 1: adaptive max-pool (W -> 32) + dense_width  =>  h_feat (B,C,H)
// One wave per (b,c,h) row: lane l owns 8 consecutive w's (one pool window).
// ---------------------------------------------------------------------------
__global__ void pool_width_kernel(const float* __restrict__ features,
                                  const float* __restrict__ w_width,
                                  const float* __restrict__ b_width,
                                  float* __restrict__ h_feat) {
    const int lane = threadIdx.x & 31;
    const int wave = threadIdx.x >> 5;
    const long base_row = ((long)blockIdx.x * 8 + wave) * 4;  // 4 rows per wave
    const float wv = w_width[lane];
    #pragma unroll
    for (int i = 0; i < 4; ++i) {
        const long r = base_row + i;                 // r < B*C*H
        const float* p = features + r * WW + lane * WIN;
        f32x4 a = __builtin_nontemporal_load((const f32x4*)p);
        f32x4 b = __builtin_nontemporal_load((const f32x4*)(p + 4));
        float m = fmaxf(fmaxf(fmaxf(a.x, a.y), fmaxf(a.z, a.w)),
                        fmaxf(fmaxf(b.x, b.y), fmaxf(b.z, b.w)));
        float val = m * wv;
        #pragma unroll
        for (int off = 16; off; off >>= 1) val += __shfl_xor(val, off, 32);
        if (lane == 0) h_feat[r] = val + b_width[0];
    }
}

// ---------------------------------------------------------------------------
// Kernel 2: per-batch small branches -> extras[b,h,a] =
//   b_enc[a] + b_cb[a] + hsum[b,a] + conv(b,:,h) . w_cb[:,a]
// Also zeroes align accumulator. One block (256 thr) per batch.
// ---------------------------------------------------------------------------
__global__ void extras_kernel(const float* __restrict__ prev,
                              const float* __restrict__ cov,
                              const float* __restrict__ hidden,
                              const float* __restrict__ conv_w,
                              const float* __restrict__ conv_b,
                              const float* __restrict__ w_cb,
                              const float* __restrict__ b_cb,
                              const float* __restrict__ w_hid,
                              const float* __restrict__ b_hid,
                              const float* __restrict__ b_enc,
                              float* __restrict__ extras,
                              float* __restrict__ align_buf) {
    __shared__ float catn[2][HH];
    __shared__ float convs[16][HH];
    __shared__ float hsum[ATT];
    __shared__ float r0[HH], r1[HH];

    const int b = blockIdx.x;
    const int t = threadIdx.x;

    if (t < HH) align_buf[b * HH + t] = 0.0f;

    // InstanceNorm1d over H per channel (population variance)
    for (int ch = 0; ch < 2; ++ch) {
        float x = 0.0f;
        if (t < HH) {
            x = (ch == 0) ? prev[b * HH + t]
                          : fminf(fmaxf(cov[b * HH + t], 0.0f), 1.0f);
            r0[t] = x;
            r1[t] = x * x;
        }
        __syncthreads();
        for (int s = 64; s >= 1; s >>= 1) {
            if (t < s) { r0[t] += r0[t + s]; r1[t] += r1[t + s]; }
            __syncthreads();
        }
        float mean = r0[0] * (1.0f / HH);
        float var  = r1[0] * (1.0f / HH) - mean * mean;
        float inv  = rsqrtf(var + 1e-5f);
        if (t < HH) catn[ch][t] = (x - mean) * inv;
        __syncthreads();
    }

    // Conv1d(2 -> 16, k=15, pad=7)
    for (int idx = t; idx < 16 * HH; idx += blockDim.x) {
        int o = idx >> 7;
        int h = idx & (HH - 1);
        float sum = conv_b[o];
        #pragma unroll
        for (int i = 0; i < 2; ++i)
            #pragma unroll
            for (int k = 0; k < 15; ++k) {
                int hh = h + k - 7;
                if (hh >= 0 && hh < HH)
                    sum += conv_w[o * 30 + i * 15 + k] * catn[i][hh];
            }
        convs[o][h] = sum;
    }

    // hidden GEMV + all scalar biases folded in (thread t == column a)
    {
        float s = b_hid[t] + b_cb[t] + b_enc[t];
        const float* hb = hidden + b * HID;
        for (int d = 0; d < HID; ++d) s += hb[d] * w_hid[d * ATT + t];
        hsum[t] = s;
    }
    __syncthreads();

    // extras[b,h,a]
    for (int idx = t; idx < HH * ATT; idx += blockDim.x) {
        int h = idx >> 8;
        int a = idx & (ATT - 1);
        float v = hsum[a];
        #pragma unroll
        for (int o = 0; o < 16; ++o) v += convs[o][h] * w_cb[o * ATT + a];
        extras[((long)(b * HH) + h) * ATT + a] = v;
    }
}

// ---------------------------------------------------------------------------
// Kernel 3: s = h_featᵀ(BH x C) @ w_enc(C x ATT) via V_WMMA_F32_16X16X4_F32,
// fused epilogue: +extras, tanh, * w_align, 16-lane reduce, atomicAdd align.
// One wave per 16x16 tile; 64 WMMAs per tile (K = 256).
// A-matrix 16x4 f32 layout: lanes0-15 M rows / K={0,1}; lanes16-31 K={2,3}.
// ---------------------------------------------------------------------------
__global__ void score_align_kernel(const float* __restrict__ h_feat,
                                   const float* __restrict__ w_enc,
                                   const float* __restrict__ extras,
                                   const float* __restrict__ w_align,
                                   float* __restrict__ align_buf) {
    const int wid  = (blockIdx.x * blockDim.x + threadIdx.x) >> 5;
    const int lane = threadIdx.x & 31;
    const int nt   = wid & 15;          // N tile (ATT/16)
    const int mt   = wid >> 4;          // M tile (B*H/16)
    const int b    = mt >> 3;
    const int h0   = (mt & 7) << 4;
    const int m    = lane & 15;         // A row / B column for this lane
    const int kh   = lane >> 4;         // K half select

    const float* Acol = h_feat + (long)b * CC * HH + (h0 + m); // + c*HH per k
    const int n0 = nt * 16;

    v8f acc = {};
    for (int k0 = 0; k0 < CC; k0 += 4) {
        const int k = k0 + kh * 2;
        v2f av, bv;
        av.x = Acol[(long)(k + 0) * HH];
        av.y = Acol[(long)(k + 1) * HH];
        bv.x = w_enc[(k + 0) * ATT + n0 + m];
        bv.y = w_enc[(k + 1) * ATT + n0 + m];
        acc = __builtin_amdgcn_wmma_f32_16x16x4_f32(
            false, av, false, bv, (short)0, acc, false, false);
    }

    // Fused epilogue: rows of this lane are v + 8*kh, column is m.
    const float wal = w_align[n0 + m];
    float partial[8];
    #pragma unroll
    for (int v = 0; v < 8; ++v) {
        const int h = h0 + v + 8 * kh;
        float sv = acc[v] + extras[((long)(b * HH) + h) * ATT + n0 + m];
        partial[v] = tanhf(sv) * wal;
    }
    // reduce over the 16 columns held by each half-wave
    #pragma unroll
    for (int v = 0; v < 8; ++v) {
        float p = partial[v];
        #pragma unroll
        for (int off = 8; off; off >>= 1) p += __shfl_xor(p, off, 32);
        partial[v] = p;
    }
    if ((lane & 15) == 0) {
        #pragma unroll
        for (int v = 0; v < 8; ++v) {
            const int h = h0 + v + 8 * kh;
            atomicAdd(&align_buf[b * HH + h], partial[v]);
        }
    }
}

// ---------------------------------------------------------------------------
// Kernel 4: softmax over H per batch. 128 threads per block.
// ---------------------------------------------------------------------------
__global__ void softmax_kernel(const float* __restrict__ align_buf,
                               const float* __restrict__ b_align,
                               float* __restrict__ attn_out) {
    __shared__ float red[HH];
    const int b = blockIdx.x;
    const int t = threadIdx.x;
    float v = align_buf[b * HH + t] + b_align[0];
    red[t] = v; __syncthreads();
    for (int s = 64; s >= 1; s >>= 1) {
        if (t < s) red[t] = fmaxf(red[t], red[t + s]);
        __syncthreads();
    }
    const float mx = red[0]; __syncthreads();
    const float e = expf(v - mx);
    red[t] = e; __syncthreads();
    for (int s = 64; s >= 1; s >>= 1) {
        if (t < s) red[t] += red[t + s];
        __syncthreads();
    }
    attn_out[b * HH + t] = e / red[0];
}

// ---------------------------------------------------------------------------
// Kernel 5: context[b,c,w] = sum_h features[b,c,h,w] * attn[b,h]
// 256 threads = 4 c-values x 64 w-quads; streaming non-temporal reads.
// ---------------------------------------------------------------------------
__global__ void context_kernel(const float* __restrict__ features,
                               const float* __restrict__ attn,
                               float* __restrict__ ctx) {
    const int blk = blockIdx.x;               // B*C/4 blocks
    const int b   = blk / (CC / 4);
    const int c   = (blk % (CC / 4)) * 4 + (threadIdx.x >> 6);
    const int wq  = (threadIdx.x & 63) * 4;
    const float* base = features + ((long)(b * CC + c)) * HH * WW + wq;
    f32x4 acc = {};
    for (int h = 0; h < HH; ++h) {
        const float a = attn[b * HH + h];     // uniform -> scalar load
        if (h + 4 < HH) __builtin_prefetch(base + (long)(h + 4) * WW, 0, 1);
        f32x4 f = __builtin_nontemporal_load((const f32x4*)(base + (long)h * WW));
        acc.x += f.x * a; acc.y += f.y * a; acc.z += f.z * a; acc.w += f.w * a;
    }
    *(f32x4*)(ctx + ((long)(b * CC + c)) * WW + wq) = acc;
}

// ---------------------------------------------------------------------------
extern "C" void kernel_launch(void* const* d_in, const int* in_sizes, int n_in,
                              void* d_out, int out_size, void* d_ws, size_t ws_size,
                              hipStream_t stream) {
    const float* features = (const float*)d_in[0];
    const float* prev     = (const float*)d_in[1];
    const float* cov      = (const float*)d_in[2];
    const float* hidden   = (const float*)d_in[3];
    const float* w_width  = (const float*)d_in[4];
    const float* b_width  = (const float*)d_in[5];
    const float* w_enc    = (const float*)d_in[6];
    const float* b_enc    = (const float*)d_in[7];
    const float* conv_w   = (const float*)d_in[8];
    const float* conv_b   = (const float*)d_in[9];
    const float* w_cb     = (const float*)d_in[10];
    const float* b_cb     = (const float*)d_in[11];
    const float* w_hid    = (const float*)d_in[12];
    const float* b_hid    = (const float*)d_in[13];
    const float* w_align  = (const float*)d_in[14];
    const float* b_align  = (const float*)d_in[15];

    float* ctx  = (float*)d_out;                         // (B,C,W)
    float* attn = ctx + (size_t)BB * CC * WW;            // (B,H)

    float* h_feat    = (float*)d_ws;                     // B*C*H
    float* extras    = h_feat + (size_t)BB * CC * HH;    // B*H*ATT
    float* align_buf = extras + (size_t)BB * HH * ATT;   // B*H

    // 1) pool + dense_width: 1,048,576 rows, 32 rows per 256-thr block
    pool_width_kernel<<<(BB * CC * HH) / 32, 256, 0, stream>>>(
        features, w_width, b_width, h_feat);

    // 2) small branches -> extras; zero align
    extras_kernel<<<BB, 256, 0, stream>>>(
        prev, cov, hidden, conv_w, conv_b, w_cb, b_cb, w_hid, b_hid, b_enc,
        extras, align_buf);

    // 3) WMMA GEMM + fused tanh/align: 256 m-tiles * 16 n-tiles, 8 waves/block
    score_align_kernel<<<(BB * HH / 16) * (ATT / 16) / 8, 256, 0, stream>>>(
        h_feat, w_enc, extras, w_align, align_buf);

    // 4) softmax over H
    softmax_kernel<<<BB, HH, 0, stream>>>(align_buf, b_align, attn);

    // 5) context pass (second 1 GiB stream)
    context_kernel<<<BB * CC / 4, 256, 0, stream>>>(features, attn, ctx);
}